// Generator_6356551598509
// MI455X (gfx1250) — compile-verified
//
#include <hip/hip_runtime.h>
#include <math.h>

typedef __attribute__((ext_vector_type(2))) float v2f;
typedef __attribute__((ext_vector_type(8))) float v8f;

#define HID 128
#define NF 32
#define NPL 8
#define RES 512
#define WAVES 2
#define PTS 32            // points per wave = two 16-row M-tiles
#define ASTRIDE 132       // 128 + pad, avoids LDS bank conflicts for both access patterns

__global__ __launch_bounds__(WAVES * 32) void Generator_6356551598509_kernel(
    const float* __restrict__ noise,      // [32,512,512]
    const float* __restrict__ coords,     // [N,2]
    const float* __restrict__ trans,      // [32,2,2]
    const float* __restrict__ input_vec,  // [128]
    const float* __restrict__ lw,         // [4,128,128]
    const float* __restrict__ lb,         // [4,128]
    const float* __restrict__ lnw,        // [4,128,8]
    const float* __restrict__ tw1,        // [128,128]
    const float* __restrict__ tb1,        // [128]
    const float* __restrict__ tw2,        // [1,128]
    const float* __restrict__ tb2,        // [1]
    float* __restrict__ out)
{
    __shared__ float act[WAVES][PTS][ASTRIDE];  // activations [point][feature]
    __shared__ float nvs[WAVES][PTS][NF];       // sampled noise values
    __shared__ float cbuf[WAVES][PTS][2];       // raw coords

    const int tid    = threadIdx.x;
    const int w      = tid >> 5;
    const int lane   = tid & 31;
    const int lane16 = lane & 15;
    const int hi     = lane >> 4;          // 0: lanes 0-15, 1: lanes 16-31
    const int pbase  = blockIdx.x * (WAVES * PTS) + w * PTS;

    // ---- Stage coords into LDS (wave-local, DS in-order => no barrier) ----
    cbuf[w][lane][0] = coords[(pbase + lane) * 2 + 0];
    cbuf[w][lane][1] = coords[(pbase + lane) * 2 + 1];
    asm volatile("s_wait_dscnt 0" ::: "memory");

    // ---- Bilinear noise sampling: lane m handles map m for all 32 points ----
    {
        const float t00 = trans[lane * 4 + 0];
        const float t01 = trans[lane * 4 + 1];
        const float t10 = trans[lane * 4 + 2];
        const float t11 = trans[lane * 4 + 3];
        const float* img = noise + (size_t)lane * RES * RES;
        for (int p = 0; p < PTS; ++p) {
            float cx = cbuf[w][p][0];
            float cy = cbuf[w][p][1];
            float xs = t00 * cx + t01 * cy - 0.5f;
            float ys = t10 * cx + t11 * cy - 0.5f;
            float x0f = floorf(xs), y0f = floorf(ys);
            float xw = xs - x0f, yw = ys - y0f;
            int x0 = ((int)x0f) & (RES - 1);
            int x1 = (x0 + 1) & (RES - 1);
            int y0 = ((int)y0f) & (RES - 1);
            int y1 = (y0 + 1) & (RES - 1);
            float i00 = img[y0 * RES + x0];
            float i01 = img[y0 * RES + x1];
            float i10 = img[y1 * RES + x0];
            float i11 = img[y1 * RES + x1];
            float i0 = i00 + (i01 - i00) * xw;
            float i1 = i10 + (i11 - i10) * xw;
            nvs[w][p][lane] = i0 + (i1 - i0) * yw;
        }
    }

    // ---- Initialize activations: x = input_vec broadcast over 32 points ----
    for (int f = lane; f < HID; f += 32) {
        float v = input_vec[f];
        for (int p = 0; p < PTS; ++p) act[w][p][f] = v;
    }
    asm volatile("s_wait_dscnt 0" ::: "memory");

    // ---- 5 GEMM layers: D[32pt,128out] = A[32,K] x B[K,128], K=136 (w/ noise) or 128
    //      Two 16-row M-tiles per wave share every B fragment (2x L2 reuse) and give
    //      two independent WMMA accumulation chains (latency hiding).
    for (int layer = 0; layer < 5; ++layer) {
        const float* W;  const float* Bz;  const float* NW = nullptr;
        int no = 0;  int withNoise;
        if (layer < 4) {
            W = lw + layer * HID * HID;  Bz = lb + layer * HID;
            NW = lnw + layer * HID * NPL; no = layer * NPL; withNoise = 1;
        } else {
            W = tw1;  Bz = tb1;  withNoise = 0;
        }

        // A fragments (register-resident for all 8 N-tiles).
        // 16x4 f32 A layout: lanes 0-15 hold (M=lane16, K=k0,k0+1); lanes 16-31 (K=k0+2,k0+3)
        v2f af0[34], af1[34];
        #pragma unroll
        for (int k = 0; k < 32; ++k) {
            af0[k] = *(const v2f*)&act[w][lane16][k * 4 + 2 * hi];
            af1[k] = *(const v2f*)&act[w][16 + lane16][k * 4 + 2 * hi];
        }
        if (withNoise) {
            af0[32] = *(const v2f*)&nvs[w][lane16][no + 2 * hi];
            af0[33] = *(const v2f*)&nvs[w][lane16][no + 4 + 2 * hi];
            af1[32] = *(const v2f*)&nvs[w][16 + lane16][no + 2 * hi];
            af1[33] = *(const v2f*)&nvs[w][16 + lane16][no + 4 + 2 * hi];
        }
        asm volatile("" ::: "memory");   // A loads issued before act is overwritten below

        for (int t = 0; t < 8; ++t) {
            const int f = t * 16 + lane16;      // output feature this lane-column owns
            v8f acc0 = {}, acc1 = {};
            const float* wr = W + f * HID + 2 * hi;   // B[k][n=f] = W[f][k]
            #pragma unroll
            for (int k = 0; k < 32; ++k) {
                v2f bf = *(const v2f*)&wr[k * 4];
                acc0 = __builtin_amdgcn_wmma_f32_16x16x4_f32(
                        false, af0[k], false, bf, (short)0, acc0, false, false);
                acc1 = __builtin_amdgcn_wmma_f32_16x16x4_f32(
                        false, af1[k], false, bf, (short)0, acc1, false, false);
            }
            if (withNoise) {
                v2f bf0 = *(const v2f*)&NW[f * NPL + 2 * hi];
                acc0 = __builtin_amdgcn_wmma_f32_16x16x4_f32(
                        false, af0[32], false, bf0, (short)0, acc0, false, false);
                acc1 = __builtin_amdgcn_wmma_f32_16x16x4_f32(
                        false, af1[32], false, bf0, (short)0, acc1, false, false);
                v2f bf1 = *(const v2f*)&NW[f * NPL + 4 + 2 * hi];
                acc0 = __builtin_amdgcn_wmma_f32_16x16x4_f32(
                        false, af0[33], false, bf1, (short)0, acc0, false, false);
                acc1 = __builtin_amdgcn_wmma_f32_16x16x4_f32(
                        false, af1[33], false, bf1, (short)0, acc1, false, false);
            }
            // Bias + leaky ReLU + writeback (C/D layout: vgpr r, lane: M=r+8*hi, N=lane16)
            const float bias = Bz[f];
            #pragma unroll
            for (int r = 0; r < 8; ++r) {
                float v0 = acc0[r] + bias;
                v0 = (v0 > 0.0f) ? v0 : 0.01f * v0;
                act[w][r + 8 * hi][f] = v0;
                float v1 = acc1[r] + bias;
                v1 = (v1 > 0.0f) ? v1 : 0.01f * v1;
                act[w][16 + r + 8 * hi][f] = v1;
            }
        }
        asm volatile("s_wait_dscnt 0" ::: "memory");
    }

    // ---- Final scalar head: out[p] = tw2 . h[p] + tb2 (one point per lane) ----
    {
        const int p = lane;
        float s = tb2[0];
        #pragma unroll 4
        for (int f = 0; f < HID; ++f) s += act[w][p][f] * tw2[f];
        out[pbase + p] = s;
    }
}

extern "C" void kernel_launch(void* const* d_in, const int* in_sizes, int n_in,
                              void* d_out, int out_size, void* d_ws, size_t ws_size,
                              hipStream_t stream) {
    const float* noise     = (const float*)d_in[0];
    const float* coords    = (const float*)d_in[1];
    const float* trans     = (const float*)d_in[2];
    const float* input_vec = (const float*)d_in[3];
    const float* lw        = (const float*)d_in[4];
    const float* lb        = (const float*)d_in[5];
    const float* lnw       = (const float*)d_in[6];
    const float* tw1       = (const float*)d_in[7];
    const float* tb1       = (const float*)d_in[8];
    const float* tw2       = (const float*)d_in[9];
    const float* tb2       = (const float*)d_in[10];

    const int n = in_sizes[1] / 2;                 // number of points
    const int blocks = n / (WAVES * PTS);          // 64 points per block
    Generator_6356551598509_kernel<<<blocks, WAVES * 32, 0, stream>>>(
        noise, coords, trans, input_vec, lw, lb, lnw, tw1, tb1, tw2, tb2,
        (float*)d_out);
}